// QDenseUndirected_old_noise_21285857919115
// MI455X (gfx1250) — compile-verified
//
#include <hip/hip_runtime.h>
#include <math.h>

#define QD 10
#define WR 10
#define NSTATE 1024

typedef __attribute__((ext_vector_type(2))) float v2f;
typedef __attribute__((ext_vector_type(8))) float v8f;

// -------- CNOT schedule as GF(2) linear maps (weight-independent, host-built) ----
struct Sched {
    unsigned short pm[QD * WR];   // stored-index pair mask  = B^{-1} e_q
    unsigned short row[QD * WR];  // selector row: parity(row & u) = basis bit q
    unsigned short bf[WR];        // final B columns: b = XOR of bf[j] over set bits of u
};

static void build_sched(Sched& sc) {
    int col[10];
    for (int j = 0; j < 10; j++) col[j] = 1 << j;   // B = identity
    for (int l = 0; l < QD; l++) {
        int rowbits[10];
        for (int p = 0; p < 10; p++) {
            int rb = 0;
            for (int j = 0; j < 10; j++) if ((col[j] >> p) & 1) rb |= 1 << j;
            rowbits[p] = rb;
        }
        int m[10], inv[10];
        for (int p = 0; p < 10; p++) { m[p] = rowbits[p]; inv[p] = 1 << p; }
        for (int cb = 0; cb < 10; cb++) {
            int piv = cb;
            while (!((m[piv] >> cb) & 1)) piv++;
            int t = m[cb]; m[cb] = m[piv]; m[piv] = t;
            t = inv[cb]; inv[cb] = inv[piv]; inv[piv] = t;
            for (int r = 0; r < 10; r++)
                if (r != cb && ((m[r] >> cb) & 1)) { m[r] ^= m[cb]; inv[r] ^= inv[cb]; }
        }
        for (int q = 0; q < WR; q++) {
            int p = 9 - q;
            sc.row[l * WR + q] = (unsigned short)rowbits[p];
            int pmv = 0;
            for (int j = 0; j < 10; j++) pmv |= ((inv[j] >> p) & 1) << j;
            sc.pm[l * WR + q] = (unsigned short)pmv;
        }
        int r = (l % (WR - 1)) + 1;
        for (int q = 0; q < WR; q++) {
            int pc = 9 - q, mt = 1 << (9 - ((q + r) % WR));
            for (int j = 0; j < 10; j++) if ((col[j] >> pc) & 1) col[j] ^= mt;
        }
    }
    for (int j = 0; j < 10; j++) sc.bf[j] = (unsigned short)col[j];
}

// -------- kernel 1: Rot gate matrices from weights -------------------------------
__global__ void gates_k(const float* __restrict__ w, float* __restrict__ g) {
    int t = threadIdx.x;
    if (t < QD * WR) {
        float phi = tanhf(w[t * 3 + 0]);
        float th  = tanhf(w[t * 3 + 1]);
        float om  = tanhf(w[t * 3 + 2]);
        float c = cosf(0.5f * th), s = sinf(0.5f * th);
        float a = 0.5f * (phi + om), b = 0.5f * (phi - om);
        float ca = cosf(a), sa = sinf(a), cb = cosf(b), sb = sinf(b);
        float* o = g + t * 8;
        o[0] =  ca * c; o[1] = -sa * c;   // g00 = e^{-ia} c
        o[2] = -cb * s; o[3] = -sb * s;   // g01 = -conj(e^{-ib}) s
        o[4] =  cb * s; o[5] = -sb * s;   // g10 = e^{-ib} s
        o[6] =  ca * c; o[7] =  sa * c;   // g11 = conj(e^{-ia}) c
    }
}

// -------- kernel 2: amplitude embedding (row-major S[n][k]) ----------------------
__global__ __launch_bounds__(256) void normalize_k(const float* __restrict__ x,
                                                   float* __restrict__ S) {
    __shared__ float red[256];
    int n = blockIdx.x, tid = threadIdx.x;
    float s = 0.f;
    for (int k = tid; k < NSTATE; k += 256) { float v = x[n * NSTATE + k]; s += v * v; }
    red[tid] = s; __syncthreads();
    for (int off = 128; off > 0; off >>= 1) {
        if (tid < off) red[tid] += red[tid + off];
        __syncthreads();
    }
    float rn = 1.0f / sqrtf(red[0]);
    for (int k = tid; k < NSTATE; k += 256) S[n * NSTATE + k] = x[n * NSTATE + k] * rn;
}

// -------- kernel 3: build full circuit unitary U (8 columns per WG, LDS-resident) -
__global__ __launch_bounds__(256) void build_u(const float* __restrict__ gates,
                                               float* __restrict__ Ur,
                                               float* __restrict__ Ui, Sched sc) {
    __shared__ float2 amp[8][NSTATE];   // 64 KB
    int wg = blockIdx.x, tid = threadIdx.x;
    for (int i = tid; i < 8 * NSTATE; i += 256) {
        int j = i >> 10, u = i & 1023;
        amp[j][u] = make_float2((u == wg * 8 + j) ? 1.0f : 0.0f, 0.0f);
    }
    for (int gi = 0; gi < QD * WR; gi++) {
        const float* g = gates + gi * 8;
        float g00r = g[0], g00i = g[1], g01r = g[2], g01i = g[3];
        float g10r = g[4], g10i = g[5], g11r = g[6], g11i = g[7];
        unsigned pm = sc.pm[gi], rw = sc.row[gi];
        unsigned pb = (unsigned)(__ffs((int)pm) - 1);
        __syncthreads();
        for (int it = 0; it < 16; ++it) {
            int idx = tid + it * 256;                  // 0..4095 = 8 cols * 512 pairs
            int j = idx >> 9;
            unsigned t = (unsigned)(idx & 511);
            unsigned u = ((t >> pb) << (pb + 1)) | (t & ((1u << pb) - 1u));
            unsigned v = u ^ pm;
            unsigned su = __popc(rw & u) & 1u;         // basis bit q of stored u
            unsigned u0 = su ? v : u;
            unsigned u1 = su ? u : v;
            float2 a0 = amp[j][u0], a1 = amp[j][u1];
            float2 n0, n1;
            n0.x = g00r * a0.x - g00i * a0.y + g01r * a1.x - g01i * a1.y;
            n0.y = g00r * a0.y + g00i * a0.x + g01r * a1.y + g01i * a1.x;
            n1.x = g10r * a0.x - g10i * a0.y + g11r * a1.x - g11i * a1.y;
            n1.y = g10r * a0.y + g10i * a0.x + g11r * a1.y + g11i * a1.x;
            amp[j][u0] = n0; amp[j][u1] = n1;
        }
    }
    __syncthreads();
    for (int i = tid; i < 8 * NSTATE; i += 256) {
        int j = i >> 10; unsigned u = (unsigned)(i & 1023);
        unsigned b = 0;
        #pragma unroll
        for (int k = 0; k < 10; k++) if ((u >> k) & 1) b ^= sc.bf[k];
        float2 a = amp[j][u];
        int c = wg * 8 + j;
        Ur[(size_t)b * NSTATE + c] = a.x;
        Ui[(size_t)b * NSTATE + c] = a.y;
    }
}

// -------- kernel 4: fp32 WMMA GEMM  C = U * S, out = clip(1024*|C|^2) ------------
#define BM 128
#define BN 64
#define BK 32
#define BKP 36            // A-tile padded stride: 16B aligned, conflict-free mod 64
#define BSTR 160          // B paired layout stride per k-pair: 160 mod 64 = 32
                          //   -> lane halves (h=0/1) hit disjoint bank halves
#define TSTR 132          // epilogue tile stride: 16B aligned float4 reads

// smem usage: lur[BM*BKP]=4608f, lui[BM*BKP]=4608f, lsp[(BK/2)*BSTR]=2560f -> 11776f
// epilogue reuses the same buffer as a 64 x TSTR tile (8448f). Total 47104 B.
#define SMEM_F 11776

__global__ __launch_bounds__(256) void gemm_prob(const float* __restrict__ Ur,
                                                 const float* __restrict__ Ui,
                                                 const float* __restrict__ S,
                                                 float* __restrict__ out) {
    __shared__ float smem[SMEM_F];
    float* lur = smem;                     // [BM][BKP]
    float* lui = smem + BM * BKP;          // [BM][BKP]
    float* lsp = smem + 2 * BM * BKP;      // [(BK/2)][BN*2] paired, stride BSTR

    int bm = blockIdx.x, bn = blockIdx.y, tid = threadIdx.x;
    int lane = tid & 31, wv = tid >> 5;
    int h = lane >> 4, nl = lane & 15;

    v8f zero;
    #pragma unroll
    for (int i = 0; i < 8; i++) zero[i] = 0.0f;
    v8f accr[4], acci[4];
    #pragma unroll
    for (int nt = 0; nt < 4; nt++) { accr[nt] = zero; acci[nt] = zero; }

    for (int k0 = 0; k0 < NSTATE; k0 += BK) {
        __syncthreads();
        // U tiles: 128 x 32, float4 global loads (coalesced), 4 per thread
        #pragma unroll
        for (int i = 0; i < 4; i++) {
            int f = tid + i * 256;
            int m = f >> 3, k4 = (f & 7) * 4;
            float4 vr = *(const float4*)(Ur + (size_t)(bm * BM + m) * NSTATE + k0 + k4);
            float4 vi = *(const float4*)(Ui + (size_t)(bm * BM + m) * NSTATE + k0 + k4);
            float* dr = &lur[m * BKP + k4];
            float* di = &lui[m * BKP + k4];
            dr[0] = vr.x; dr[1] = vr.y; dr[2] = vr.z; dr[3] = vr.w;
            di[0] = vi.x; di[1] = vi.y; di[2] = vi.z; di[3] = vi.w;
        }
        // S tile: 64 rows (n) x 32 (k) -> paired layout lsp[k/2][n*2 + (k&1)]
        #pragma unroll
        for (int i = 0; i < 2; i++) {
            int f = tid + i * 256;
            int n = f >> 3, k4 = (f & 7) * 4;
            float4 v = *(const float4*)(S + (size_t)(bn * BN + n) * NSTATE + k0 + k4);
            float2* p0 = (float2*)&lsp[(k4 >> 1) * BSTR + n * 2];
            float2* p1 = (float2*)&lsp[((k4 >> 1) + 1) * BSTR + n * 2];
            *p0 = make_float2(v.x, v.y);
            *p1 = make_float2(v.z, v.w);
        }
        __syncthreads();

        int arow = (wv * 16 + nl) * BKP;
        #pragma unroll
        for (int kk = 0; kk < BK; kk += 4) {
            // A fragments: K = kk + 2h + j  (contiguous pair, 8B-aligned)
            float2 tr = *(const float2*)&lur[arow + kk + 2 * h];
            float2 ti = *(const float2*)&lui[arow + kk + 2 * h];
            v2f ar;  ar.x  = tr.x; ar.y  = tr.y;
            v2f ai2; ai2.x = ti.x; ai2.y = ti.y;
            #pragma unroll
            for (int nt = 0; nt < 4; nt++) {
                // B fragment: one 8B LDS load, no repacking
                float2 tb = *(const float2*)&lsp[(kk / 2 + h) * BSTR + (nt * 16 + nl) * 2];
                v2f b; b.x = tb.x; b.y = tb.y;
                accr[nt] = __builtin_amdgcn_wmma_f32_16x16x4_f32(
                    false, ar, false, b, (short)0, accr[nt], false, false);
                acci[nt] = __builtin_amdgcn_wmma_f32_16x16x4_f32(
                    false, ai2, false, b, (short)0, acci[nt], false, false);
            }
        }
    }

    // ---- epilogue: |.|^2, clip, stage through LDS, coalesced float4 stores ----
    __syncthreads();
    float* tile = smem;                    // [BN][TSTR]
    int mcol = wv * 16 + h * 8;
    #pragma unroll
    for (int nt = 0; nt < 4; nt++) {
        int nrow = nt * 16 + nl;
        #pragma unroll
        for (int v = 0; v < 8; v++) {
            float cr = accr[nt][v], ci = acci[nt][v];
            float p = (cr * cr + ci * ci) * 1024.0f;
            p = fminf(fmaxf(p, 0.0f), 1.0f);
            tile[nrow * TSTR + mcol + v] = p;
        }
    }
    __syncthreads();
    #pragma unroll
    for (int i = 0; i < 8; i++) {
        int f = tid + i * 256;
        int n = f >> 5, m4 = (f & 31) * 4;
        float4 v = *(const float4*)&tile[n * TSTR + m4];
        *(float4*)(out + (size_t)(bn * BN + n) * NSTATE + bm * BM + m4) = v;
    }
}

// -------- launcher ---------------------------------------------------------------
extern "C" void kernel_launch(void* const* d_in, const int* in_sizes, int n_in,
                              void* d_out, int out_size, void* d_ws, size_t ws_size,
                              hipStream_t stream) {
    const float* x = (const float*)d_in[0];       // [1024,1,32,32]
    const float* w = (const float*)d_in[1];       // [10,10,3]
    float* out = (float*)d_out;                   // [1024,1,32,32]
    float* ws  = (float*)d_ws;

    float* gates = ws;                            // 800 floats (pad to 1024)
    float* S     = ws + 1024;                     // 1M floats
    float* Ur    = S + (size_t)NSTATE * NSTATE;   // 1M floats
    float* Ui    = Ur + (size_t)NSTATE * NSTATE;  // 1M floats

    Sched sc;
    build_sched(sc);

    gates_k    <<<1, 128, 0, stream>>>(w, gates);
    normalize_k<<<NSTATE, 256, 0, stream>>>(x, S);
    build_u    <<<NSTATE / 8, 256, 0, stream>>>(gates, Ur, Ui, sc);
    dim3 g(NSTATE / BM, NSTATE / BN);
    gemm_prob  <<<g, 256, 0, stream>>>(Ur, Ui, S, out);
}